// PointNetSetAbstraction_49898930045497
// MI455X (gfx1250) — compile-verified
//
#include <hip/hip_runtime.h>
#include <hip/hip_bf16.h>

#define B_   8
#define N_   4096
#define S_   1024
#define K_   32
#define M_   (B_*S_*K_)   /* 262144 rows through the MLP */
#define RAD2 0.16f
#define EPS_ 1e-5f

typedef __attribute__((ext_vector_type(2))) float v2f;
typedef __attribute__((ext_vector_type(8))) float v8f;

// ---------------------------------------------------------------------------
// 1) Farthest-point sampling: one workgroup per batch, LDS distance array,
//    block-wide argmax with first-index tie-break. Replicates the reference's
//    int32-truncation quirk on per-component squared distances.
// ---------------------------------------------------------------------------
__global__ void fps_kernel(const float* __restrict__ xyz,      // (B,3,N)
                           const int*   __restrict__ farthest_init,
                           int*   __restrict__ fps_idx,        // (B,S)
                           float* __restrict__ nxyz_ws,        // (B,S,3)
                           float* __restrict__ out_xyz) {      // (B,3,S)
  __shared__ float dist[N_];
  __shared__ float rval[256];
  __shared__ int   ridx[256];
  __shared__ int   far_s;

  const int b = blockIdx.x;
  const int t = threadIdx.x;
  const float* xb = xyz + (size_t)b * 3 * N_;

  for (int n = t; n < N_; n += 256) dist[n] = 1e10f;
  if (t == 0) far_s = farthest_init[b];
  __syncthreads();

  for (int i = 0; i < S_; ++i) {
    const int far = far_s;
    if (t == 0) fps_idx[b * S_ + i] = far;
    const float cx = xb[far], cy = xb[N_ + far], cz = xb[2 * N_ + far];

    float bv = -1e30f; int bi = 0;
    for (int n = t; n < N_; n += 256) {
      float dx = xb[n] - cx, dy = xb[N_ + n] - cy, dz = xb[2 * N_ + n] - cz;
      float d = (float)((int)(dx * dx) + (int)(dy * dy) + (int)(dz * dz));
      float dn = fminf(dist[n], d);
      dist[n] = dn;
      if (dn > bv) { bv = dn; bi = n; }   // strict > keeps earliest index per thread
    }
    rval[t] = bv; ridx[t] = bi;
    __syncthreads();
    for (int sft = 128; sft > 0; sft >>= 1) {
      if (t < sft) {
        if (rval[t + sft] > rval[t] ||
            (rval[t + sft] == rval[t] && ridx[t + sft] < ridx[t])) {
          rval[t] = rval[t + sft]; ridx[t] = ridx[t + sft];
        }
      }
      __syncthreads();
    }
    if (t == 0) far_s = ridx[0];
    __syncthreads();
  }

  // gather new_xyz: both transposed output (B,3,S) and row-major ws (B,S,3)
  for (int s = t; s < S_; s += 256) {
    int idx = fps_idx[b * S_ + s];
    float x = xb[idx], y = xb[N_ + idx], z = xb[2 * N_ + idx];
    out_xyz[(size_t)(b * 3 + 0) * S_ + s] = x;
    out_xyz[(size_t)(b * 3 + 1) * S_ + s] = y;
    out_xyz[(size_t)(b * 3 + 2) * S_ + s] = z;
    float* nr = nxyz_ws + (size_t)(b * S_ + s) * 3;
    nr[0] = x; nr[1] = y; nr[2] = z;
  }
}

// ---------------------------------------------------------------------------
// 2) Ball query + grouping: one wave32 per center. Scan points in index order,
//    ballot + prefix popcount assigns the first 32 hits (d <= r^2) to slots;
//    slots beyond the hit count are padded with the first hit. Each lane then
//    emits one grouped feature row [gx,gy,gz,px,py,pz,0,0] (Cin padded to 8).
// ---------------------------------------------------------------------------
__global__ void query_group_kernel(const float* __restrict__ xyz,    // (B,3,N)
                                   const float* __restrict__ pts,    // (B,3,N)
                                   const float* __restrict__ nxyz_ws,// (B,S,3)
                                   float* __restrict__ X0) {         // (M,8)
  __shared__ int idxbuf[8][K_];
  const int wave = threadIdx.x >> 5;
  const int lane = threadIdx.x & 31;
  const int g = blockIdx.x * 8 + wave;       // global center id
  const int b = g >> 10;
  const float* xb = xyz + (size_t)b * 3 * N_;

  const float cx = nxyz_ws[(size_t)g * 3 + 0];
  const float cy = nxyz_ws[(size_t)g * 3 + 1];
  const float cz = nxyz_ws[(size_t)g * 3 + 2];
  const float cn = cx * cx + cy * cy + cz * cz;

  int filled = 0;
  for (int j = 0; j < N_ && filled < K_; j += 32) {
    const int n = j + lane;
    float x = xb[n], y = xb[N_ + n], z = xb[2 * N_ + n];
    float d = -2.0f * (cx * x + cy * y + cz * z) + cn + (x * x + y * y + z * z);
    bool cond = !(d > RAD2);
    unsigned mask = (unsigned)__ballot(cond);
    int rank = __popc(mask & ((1u << lane) - 1u));
    int slot = filled + rank;
    if (cond && slot < K_) idxbuf[wave][slot] = n;
    filled += __popc(mask);
  }
  __syncthreads();

  const int kcount = filled < K_ ? filled : K_;
  const int kidx = (lane < kcount) ? idxbuf[wave][lane] : idxbuf[wave][0];

  const float* pb = pts + (size_t)b * 3 * N_;
  float gx = xb[kidx] - cx, gy = xb[N_ + kidx] - cy, gz = xb[2 * N_ + kidx] - cz;
  float px = pb[kidx], py = pb[N_ + kidx], pz = pb[2 * N_ + kidx];

  float* row = X0 + ((size_t)g * K_ + lane) * 8;
  row[0] = gx; row[1] = gy; row[2] = gz;
  row[3] = px; row[4] = py; row[5] = pz;
  row[6] = 0.0f; row[7] = 0.0f;
}

// ---------------------------------------------------------------------------
// 3a) Pad weight matrix (Cout x cin) into (Cout x kpad) with zeros so the
//     GEMM kernel needs no bounds checks (removes EXEC-masked branchy loads).
// ---------------------------------------------------------------------------
__global__ void pad_w_kernel(const float* __restrict__ W, int cin,
                             float* __restrict__ Wp, int kpad, int total) {
  const int i = blockIdx.x * blockDim.x + threadIdx.x;
  if (i >= total) return;
  const int o = i / kpad, k = i % kpad;
  Wp[i] = (k < cin) ? W[o * cin + k] : 0.0f;
}

// ---------------------------------------------------------------------------
// 3b) GEMM tile kernel: one wave32 per 16x16 output tile of Y = X * Wp^T + b.
//     KPAD is a compile-time constant -> fully unrolled chain of
//     V_WMMA_F32_16X16X4_F32 with unguarded b64 fragment loads.
//     A frag layout: lane<16 -> K=kk..kk+1 ; lane>=16 -> K=kk+2..kk+3.
//     C/D layout: VGPR r -> rows r / r+8, col = lane%16.
// ---------------------------------------------------------------------------
template <int KPAD>
__global__ void gemm_wmma_kernel(const float* __restrict__ X, int ldx,
                                 const float* __restrict__ Wp,
                                 const float* __restrict__ bias,
                                 float* __restrict__ Y, int ldy) {
  const int lane = threadIdx.x;
  const int row0 = blockIdx.x << 4;
  const int n0   = blockIdx.y << 4;
  const int m    = lane & 15;
  const int koff = (lane >> 4) << 1;

  const float* xrow = X  + (size_t)(row0 + m) * ldx  + koff;
  const float* wrow = Wp + (size_t)(n0 + m)  * KPAD + koff;

  v8f c = {};
  #pragma unroll
  for (int kk = 0; kk < KPAD; kk += 4) {
    v2f a = *(const v2f*)(xrow + kk);
    v2f b = *(const v2f*)(wrow + kk);
    c = __builtin_amdgcn_wmma_f32_16x16x4_f32(
        /*neg_a=*/false, a, /*neg_b=*/false, b,
        /*c_mod=*/(short)0, c, /*reuse_a=*/false, /*reuse_b=*/false);
  }

  const float bv = bias[n0 + m];
  const int rbase = row0 + ((lane >> 4) << 3);
  #pragma unroll
  for (int r = 0; r < 8; ++r)
    Y[(size_t)(rbase + r) * ldy + (n0 + m)] = c[r] + bv;
}

// ---------------------------------------------------------------------------
// 4) Per-channel sum / sum-of-squares for batch-norm statistics.
// ---------------------------------------------------------------------------
__global__ void colsum_kernel(const float* __restrict__ Y, int ldy, int rows,
                              float* __restrict__ sum, float* __restrict__ sq) {
  __shared__ float ssum[256];
  __shared__ float ssq[256];
  const int col = blockIdx.x;
  const int t = threadIdx.x;
  float a = 0.0f, q = 0.0f;
  for (int r = t; r < rows; r += 256) {
    float v = Y[(size_t)r * ldy + col];
    a += v; q += v * v;
  }
  ssum[t] = a; ssq[t] = q;
  __syncthreads();
  for (int sft = 128; sft > 0; sft >>= 1) {
    if (t < sft) { ssum[t] += ssum[t + sft]; ssq[t] += ssq[t + sft]; }
    __syncthreads();
  }
  if (t == 0) { sum[col] = ssum[0]; sq[col] = ssq[0]; }
}

// ---------------------------------------------------------------------------
// 5) BN (batch statistics) + ReLU, in place.
// ---------------------------------------------------------------------------
__global__ void bnrelu_kernel(float* __restrict__ Y, int cout, int rows,
                              const float* __restrict__ sum,
                              const float* __restrict__ sq,
                              const float* __restrict__ g,
                              const float* __restrict__ beta) {
  const size_t total = (size_t)rows * cout;
  const float invM = 1.0f / (float)rows;
  for (size_t i = (size_t)blockIdx.x * blockDim.x + threadIdx.x; i < total;
       i += (size_t)gridDim.x * blockDim.x) {
    int c = (int)(i % cout);
    float mu = sum[c] * invM;
    float var = sq[c] * invM - mu * mu;
    float v = (Y[i] - mu) * rsqrtf(var + EPS_) * g[c] + beta[c];
    Y[i] = fmaxf(v, 0.0f);
  }
}

// ---------------------------------------------------------------------------
// 6) Max over the K=32 sample dimension; writes transposed (B,Cout,S).
// ---------------------------------------------------------------------------
__global__ void maxpool_kernel(const float* __restrict__ Y,   // (M,128)
                               float* __restrict__ out) {     // (B,128,S)
  const int i = blockIdx.x * blockDim.x + threadIdx.x;  // < B*128*S
  const int b = i >> 17;                 // / (128*1024)
  const int o = (i >> 10) & 127;
  const int s = i & 1023;
  const float* base = Y + ((size_t)(b * S_ + s) * K_) * 128 + o;
  float m = base[0];
  #pragma unroll
  for (int k = 1; k < K_; ++k) m = fmaxf(m, base[(size_t)k * 128]);
  out[i] = m;
}

__global__ void zero_kernel(float* __restrict__ p, int n) {
  int i = blockIdx.x * blockDim.x + threadIdx.x;
  if (i < n) p[i] = 0.0f;
}

// ---------------------------------------------------------------------------
extern "C" void kernel_launch(void* const* d_in, const int* in_sizes, int n_in,
                              void* d_out, int out_size, void* d_ws, size_t ws_size,
                              hipStream_t stream) {
  const float* xyz   = (const float*)d_in[0];
  const float* pts   = (const float*)d_in[1];
  const int*   finit = (const int*)d_in[2];
  const float* W0 = (const float*)d_in[3],  *b0 = (const float*)d_in[4];
  const float* g0 = (const float*)d_in[5],  *be0 = (const float*)d_in[6];
  const float* W1 = (const float*)d_in[7],  *b1 = (const float*)d_in[8];
  const float* g1 = (const float*)d_in[9],  *be1 = (const float*)d_in[10];
  const float* W2 = (const float*)d_in[11], *b2 = (const float*)d_in[12];
  const float* g2 = (const float*)d_in[13], *be2 = (const float*)d_in[14];

  float* out_xyz = (float*)d_out;                    // (B,3,S)
  float* out_pts = out_xyz + (size_t)B_ * 3 * S_;    // (B,128,S)

  char* w = (char*)d_ws;
  auto take = [&](size_t bytes) {
    char* p = w;
    w += (bytes + 255) & ~(size_t)255;
    return p;
  };
  int*   fps_idx = (int*)  take((size_t)B_ * S_ * sizeof(int));
  float* nxyz    = (float*)take((size_t)B_ * S_ * 3 * sizeof(float));
  float* sums    = (float*)take(512 * sizeof(float));   // [s0|q0|s1|q1|s2(128)|q2(128)]
  float* Wp0     = (float*)take(64 * 8 * sizeof(float));
  float* X0      = (float*)take((size_t)M_ * 8   * sizeof(float));
  float* R1      = (float*)take((size_t)M_ * 128 * sizeof(float));
  float* R2      = (float*)take((size_t)M_ * 128 * sizeof(float));

  // 1) FPS + gather centers
  fps_kernel<<<B_, 256, 0, stream>>>(xyz, finit, fps_idx, nxyz, out_xyz);

  // 2) ball query + grouped feature rows (M x 8, zero-padded cin)
  query_group_kernel<<<(B_ * S_) / 8, 256, 0, stream>>>(xyz, pts, nxyz, X0);

  // BN statistic buffers + padded layer-0 weights
  zero_kernel<<<2, 256, 0, stream>>>(sums, 512);
  pad_w_kernel<<<2, 256, 0, stream>>>(W0, 6, Wp0, 8, 64 * 8);

  // 3) layer 1: (M,8) x (64,8) -> R1 (ld 64)
  gemm_wmma_kernel<8><<<dim3(M_ / 16, 4), 32, 0, stream>>>(X0, 8, Wp0, b0, R1, 64);
  colsum_kernel<<<64, 256, 0, stream>>>(R1, 64, M_, sums + 0, sums + 64);
  bnrelu_kernel<<<2048, 256, 0, stream>>>(R1, 64, M_, sums + 0, sums + 64, g0, be0);

  // 4) layer 2: (M,64) x (64,64) -> R2 (ld 64); W1 already dense 64x64
  gemm_wmma_kernel<64><<<dim3(M_ / 16, 4), 32, 0, stream>>>(R1, 64, W1, b1, R2, 64);
  colsum_kernel<<<64, 256, 0, stream>>>(R2, 64, M_, sums + 128, sums + 192);
  bnrelu_kernel<<<2048, 256, 0, stream>>>(R2, 64, M_, sums + 128, sums + 192, g1, be1);

  // 5) layer 3: (M,64) x (128,64) -> R1 (ld 128); W2 dense 128x64
  gemm_wmma_kernel<64><<<dim3(M_ / 16, 8), 32, 0, stream>>>(R2, 64, W2, b2, R1, 128);
  colsum_kernel<<<128, 256, 0, stream>>>(R1, 128, M_, sums + 256, sums + 384);
  bnrelu_kernel<<<4096, 256, 0, stream>>>(R1, 128, M_, sums + 256, sums + 384, g2, be2);

  // 6) max over K, transposed store
  maxpool_kernel<<<(B_ * 128 * S_) / 256, 256, 0, stream>>>(R1, out_pts);
}